// DifferentiableNeuralDictionary_71734543778597
// MI455X (gfx1250) — compile-verified
//
#include <hip/hip_runtime.h>
#include <hip/hip_bf16.h>
#include <cstdint>

typedef __attribute__((ext_vector_type(2))) float v2f;
typedef __attribute__((ext_vector_type(8))) float v8f;

#define A_DIM   8
#define N_KEYS  50000
#define D_DIM   64
#define B_DIM   256
#define K_NN    50
#define QB      16      // query rows per workgroup
#define TILE    128     // keys per LDS tile (two 16-col slabs per wave)
#define CAP     128     // candidate buffer per row (>= TILE: no overflow per tile)
#define DELTA   0.001f
#define BIGF    3.0e38f
#define NTHREADS 128

// d_out flat offsets (floats)
#define OFF_MAXV   0
#define OFF_ACT    256
#define OFF_VALS   512
#define OFF_IDX    2560
#define OFF_SCORE  104960

// ---- gfx1250 async global->LDS path (graceful fallback if builtin absent) ----
#if defined(__AMDGCN__) && __has_builtin(__builtin_amdgcn_global_load_async_to_lds_b128) && __has_builtin(__builtin_amdgcn_s_wait_asynccnt)
#define USE_ASYNC_LDS 1
#else
#define USE_ASYNC_LDS 0
#endif

// builtin's param type (from hipcc diagnostic): pointer to 16B int vector
typedef int v4i_t __attribute__((vector_size(4 * sizeof(int))));
typedef __attribute__((address_space(3))) v4i_t lds_v4i;

// Stage one TILE x D key tile into an LDS buffer. 2048 float4 chunks / 128 thr.
__device__ __forceinline__ void stage_tile(const float* __restrict__ kbase,
                                           int t0, float (*Kt)[D_DIM], int tid)
{
    const int remain = N_KEYS - t0;
    const int valid  = remain < TILE ? remain : TILE;
    #pragma unroll
    for (int j = 0; j < (TILE * (D_DIM / 4)) / NTHREADS; ++j) {
        int i   = tid + NTHREADS * j;
        int row = i >> 4;                       // 16 float4 per key row
        float4* dst = ((float4*)&Kt[0][0]) + i;
        if (row < valid) {
            const float* src = kbase + (size_t)(t0 + row) * D_DIM + (size_t)(i & 15) * 4;
#if USE_ASYNC_LDS
            __builtin_amdgcn_global_load_async_to_lds_b128(
                (v4i_t*)src,
                (lds_v4i*)(uint32_t)(uintptr_t)dst,
                /*offset=*/0, /*cpol=*/0);
#else
            *dst = *(const float4*)src;
#endif
        } else {
            *dst = make_float4(0.f, 0.f, 0.f, 0.f); // tail padding
        }
    }
}

__device__ __forceinline__ void wait_tile()
{
#if USE_ASYNC_LDS
    __builtin_amdgcn_s_wait_asynccnt(0);
#endif
    __syncthreads();
}

__global__ __launch_bounds__(NTHREADS) void dnd_main(
    const float* __restrict__ query,      // [B, D]
    const float* __restrict__ mem_keys,   // [A, N, D]
    const float* __restrict__ mem_values, // [A, N]
    float* __restrict__ out)
{
    __shared__ float Q[QB][D_DIM];          // raw queries (for exact recompute)
    __shared__ float qsq[QB];
    __shared__ float Kt[2][TILE][D_DIM];    // double-buffered key tiles (64 KB)
    __shared__ float ksq[TILE];
    __shared__ float topd[QB][K_NN];        // sorted ascending
    __shared__ int   topi[QB][K_NN];
    __shared__ float thr[QB];               // current 50th-best
    __shared__ float cand_d[QB][CAP];
    __shared__ int   cand_i[QB][CAP];
    __shared__ int   cand_cnt[QB];
    __shared__ float wbuf[QB][K_NN];
    __shared__ float wvbuf[QB][K_NN];

    const int tid  = threadIdx.x;
    const int lane = tid & 31;
    const int wave = tid >> 5;
    const int a    = blockIdx.x >> 4;   // action (8)
    const int qb   = blockIdx.x & 15;   // query block (16)
    const int b0   = qb * QB;

    const float* qbase = query + (size_t)b0 * D_DIM;
    const float* kbase = mem_keys + (size_t)a * N_KEYS * D_DIM;
    const float* vbase = mem_values + (size_t)a * N_KEYS;

    // ---- stage query block into LDS ----
    for (int i = tid; i < QB * (D_DIM / 4); i += NTHREADS) {
        ((float4*)&Q[0][0])[i] = ((const float4*)qbase)[i];
    }
    for (int i = tid; i < QB * K_NN; i += NTHREADS) {
        (&topd[0][0])[i] = BIGF;
        (&topi[0][0])[i] = 0;
    }
    if (tid < QB) { cand_cnt[tid] = 0; thr[tid] = BIGF; }
    __syncthreads();
    if (tid < QB) {
        float s = 0.f;
        for (int d = 0; d < D_DIM; ++d) { float x = Q[tid][d]; s += x * x; }
        qsq[tid] = s;
    }
    __syncthreads();

    // ---- preload A fragments (A = -2*Q): wmma f32 16x16x4 layout ----
    // lane<16: M=lane, K=4k+{0,1}; lane>=16: M=lane-16, K=4k+{2,3}
    const int m    = lane & 15;
    const int kb   = (lane >> 4) * 2;
    const int n    = lane & 15;
    const int half = lane >> 4;
    v2f afrag[16];
    #pragma unroll
    for (int kk = 0; kk < 16; ++kk) {
        afrag[kk].x = -2.0f * Q[m][4 * kk + kb + 0];
        afrag[kk].y = -2.0f * Q[m][4 * kk + kb + 1];
    }
    // preload per-lane row norms (C/D row = j + 8*half)
    float qsq_r[8];
    #pragma unroll
    for (int j = 0; j < 8; ++j) qsq_r[j] = qsq[j + 8 * half];

    // ================= main scan over N (double-buffered) =================
    const int ntiles = (N_KEYS + TILE - 1) / TILE;
    stage_tile(kbase, 0, Kt[0], tid);

    for (int ti = 0; ti < ntiles; ++ti) {
        const int t0  = ti * TILE;
        const int buf = ti & 1;
        const int remain = N_KEYS - t0;
        const int valid  = remain < TILE ? remain : TILE;

        wait_tile();                       // current tile resident in LDS
        if (ti + 1 < ntiles)               // prefetch next tile into other buffer
            stage_tile(kbase, t0 + TILE, Kt[buf ^ 1], tid);

        // per-key squared norms (one row per thread; TILE == NTHREADS)
        {
            float s;
            if (tid < valid) {
                s = 0.f;
                const float* kr = &Kt[buf][tid][0];
                for (int d = 0; d < D_DIM; ++d) { float x = kr[d]; s += x * x; }
            } else {
                s = BIGF;                  // padded columns never selected
            }
            ksq[tid] = s;
        }
        __syncthreads();

        // ---- two 16-column slabs per wave: chained fp32 WMMAs ----
        for (int slab = 0; slab < 2; ++slab) {
            const int cbase = slab * 64 + wave * 16;
            v8f c = {};
            #pragma unroll
            for (int kk = 0; kk < 16; ++kk) {
                v2f bfrag;
                bfrag.x = Kt[buf][cbase + m][4 * kk + kb + 0];
                bfrag.y = Kt[buf][cbase + m][4 * kk + kb + 1];
                c = __builtin_amdgcn_wmma_f32_16x16x4_f32(
                        false, afrag[kk], false, bfrag,
                        (short)0, c, false, false);
            }

            // threshold filter + candidate append
            const int   coll = cbase + n;
            const float kcol = ksq[coll];
            #pragma unroll
            for (int j = 0; j < 8; ++j) {
                int   row = j + 8 * half;
                float d2  = qsq_r[j] + kcol + c[j];
                if (d2 < thr[row]) {
                    int pos = atomicAdd(&cand_cnt[row], 1);
                    if (pos < CAP) { cand_d[row][pos] = d2; cand_i[row][pos] = t0 + coll; }
                }
            }
        }
        __syncthreads();

        // ---- serial merge, one thread per query row ----
        if (tid < QB) {
            int cnt = cand_cnt[tid]; if (cnt > CAP) cnt = CAP;
            for (int ci = 0; ci < cnt; ++ci) {
                float d = cand_d[tid][ci];
                if (d < topd[tid][K_NN - 1]) {
                    int idx = cand_i[tid][ci];
                    int p = K_NN - 1;
                    while (p > 0 && topd[tid][p - 1] > d) {
                        topd[tid][p] = topd[tid][p - 1];
                        topi[tid][p] = topi[tid][p - 1];
                        --p;
                    }
                    topd[tid][p] = d; topi[tid][p] = idx;
                }
            }
            cand_cnt[tid] = 0;
            thr[tid] = topd[tid][K_NN - 1];
        }
        __syncthreads();
    }

    // ================= exact recompute + outputs =================
    for (int t = tid; t < QB * K_NN; t += NTHREADS) {
        int r = t / K_NN, k = t % K_NN;
        int idx = topi[r][k];
        const float* kr = kbase + (size_t)idx * D_DIM;
        float s = 0.f;
        for (int d = 0; d < D_DIM; ++d) {
            float diff = Q[r][d] - kr[d];
            s += diff * diff;
        }
        float w  = 1.0f / (s + DELTA);
        float gv = vbase[idx];
        wbuf[r][k]  = w;
        wvbuf[r][k] = w * gv;
        int b = b0 + r;
        size_t pair = ((size_t)b * A_DIM + a) * K_NN + k;
        out[OFF_IDX + pair]   = (float)idx;
        out[OFF_SCORE + pair] = s;
    }
    __syncthreads();
    if (tid < QB) {
        float sw = 0.f, swv = 0.f;
        for (int k = 0; k < K_NN; ++k) { sw += wbuf[tid][k]; swv += wvbuf[tid][k]; }
        int b = b0 + tid;
        out[OFF_VALS + (size_t)b * A_DIM + a] = swv / sw;
    }
}

__global__ __launch_bounds__(256) void dnd_reduce(float* __restrict__ out)
{
    int b = blockIdx.x * blockDim.x + threadIdx.x;
    if (b >= B_DIM) return;
    float best = out[OFF_VALS + (size_t)b * A_DIM];
    int bi = 0;
    for (int a2 = 1; a2 < A_DIM; ++a2) {
        float v = out[OFF_VALS + (size_t)b * A_DIM + a2];
        if (v > best) { best = v; bi = a2; }  // first occurrence wins, like jnp.argmax
    }
    out[OFF_MAXV + b] = best;
    out[OFF_ACT + b]  = (float)bi;
}

extern "C" void kernel_launch(void* const* d_in, const int* in_sizes, int n_in,
                              void* d_out, int out_size, void* d_ws, size_t ws_size,
                              hipStream_t stream) {
    const float* query      = (const float*)d_in[0];
    const float* mem_keys   = (const float*)d_in[1];
    const float* mem_values = (const float*)d_in[2];
    float* out = (float*)d_out;

    dnd_main<<<A_DIM * (B_DIM / QB), NTHREADS, 0, stream>>>(query, mem_keys, mem_values, out);
    dnd_reduce<<<1, 256, 0, stream>>>(out);
}